// DeepIRT_15942918603386
// MI455X (gfx1250) — compile-verified
//
#include <hip/hip_runtime.h>
#include <math.h>

// DeepIRT fused forward for MI455X (gfx1250, wave32).
// One workgroup per batch element; everything resident in LDS (~266 KB / 320 KB).
// All GEMMs use exact-precision V_WMMA_F32_16X16X4_F32 (fp32 WMMA).

#define NUM_Q  10000
#define NUM_C  300
#define DIM_S  64
#define SIZE_M 50
#define BATCH  128
#define TT     200
#define HALF   32          // embedding half-width

#define RK  212            // sK rows (need up to row 208 for k[t'+1] in f-GEMM; zero-padded)
#define RP  208            // padded rows (13 WMMA M-tiles of 16)
#define WST 52             // sW row stride (50 padded)

typedef __attribute__((ext_vector_type(2))) float v2f;
typedef __attribute__((ext_vector_type(8))) float v8f;

static __device__ __forceinline__ v8f wmma_f32_k4(v2f a, v2f b, v8f c) {
  // D = A(16x4,f32) x B(4x16,f32) + C(16x16,f32); 8-arg VOP3P form
  return __builtin_amdgcn_wmma_f32_16x16x4_f32(false, a, false, b, (short)0, c, false, false);
}

__global__ __launch_bounds__(256) void deepirt_fused_kernel(
    const int*   __restrict__ question, const int*   __restrict__ response,
    const float* __restrict__ vq_emb,   const float* __restrict__ vc_emb,
    const float* __restrict__ kq_emb,   const float* __restrict__ kc_emb,
    const float* __restrict__ Mk,       const float* __restrict__ Mv0,
    const float* __restrict__ eW, const float* __restrict__ eb,
    const float* __restrict__ aW, const float* __restrict__ ab,
    const float* __restrict__ fW, const float* __restrict__ fb,
    const float* __restrict__ abilW, const float* __restrict__ abilb,
    const float* __restrict__ diffW, const float* __restrict__ diffb,
    float* __restrict__ out)
{
  extern __shared__ float smem[];
  float* sK   = smem;                 // [RK][64]   k rows (pad rows zero)
  float* sV   = sK  + RK * 64;        // [RP][64]   v rows; later reused as read[t']
  float* sE   = sV  + RP * 64;        // [RP][64]   e rows; later reused as f rows
  float* sA   = sE  + RP * 64;        // [RP][64]   a rows
  float* sW   = sA  + RP * 64;        // [RP][WST]  softmax weights (cols 0..49)
  float* sMv  = sW  + RP * WST;       // [50][64]   value memory
  float* sRed = sMv + SIZE_M * 64;    // [4][64]    read-reduction scratch
  int*   sQ   = (int*)(sRed + 4 * 64);   // [TT]
  int*   sRsp = sQ + TT;                 // [TT]

  const int b    = blockIdx.x;
  const int tid  = threadIdx.x;
  const int lane = tid & 31;
  const int wave = tid >> 5;
  const int hh   = lane >> 4;     // which half of the wave (0/1)
  const int r16  = lane & 15;     // M (A) / N (B,C) index within tile

  // ---- Phase 0: cache per-t indices -----------------------------------
  for (int t = tid; t < TT; t += 256) {
    sQ[t]   = question[b * TT + t];
    sRsp[t] = response[b * TT + t];
  }
  __syncthreads();

  // ---- Phase 1: gather k, v into LDS (zero padding rows); init Mv -----
  for (int idx = tid; idx < RK * 64; idx += 256) {
    int t = idx >> 6, c = idx & 63;
    float x = 0.f;
    if (t < TT) {
      int q = sQ[t];
      x = (c < HALF) ? kq_emb[q * HALF + c] : kc_emb[c - HALF];   // skill == 0
    }
    sK[idx] = x;
  }
  for (int idx = tid; idx < RP * 64; idx += 256) {
    int t = idx >> 6, c = idx & 63;
    float x = 0.f;
    if (t < TT) {
      int q = sQ[t], r = sRsp[t];
      x = (c < HALF) ? vq_emb[(q + NUM_Q * r) * HALF + c]
                     : vc_emb[(NUM_C * r) * HALF + (c - HALF)];
    }
    sV[idx] = x;
  }
  for (int idx = tid; idx < SIZE_M * 64; idx += 256) sMv[idx] = Mv0[idx];
  __syncthreads();

  // ---- Phase 2a: wlog = k @ Mk^T (N: 50 padded to 64), WMMA f32 K=4 ---
  for (int nt = 0; nt < 4; ++nt) {
    int o  = nt * 16 + r16;
    int oc = (o < SIZE_M) ? o : (SIZE_M - 1);   // clamp pad cols (discarded)
    v2f bf[16];
#pragma unroll
    for (int kb = 0; kb < 16; ++kb)
      bf[kb] = *(const v2f*)&Mk[oc * 64 + kb * 4 + 2 * hh];
    for (int mt = wave; mt < RP / 16; mt += 8) {
      v2f af[16];
#pragma unroll
      for (int kb = 0; kb < 16; ++kb)
        af[kb] = *(const v2f*)&sK[(mt * 16 + r16) * 64 + kb * 4 + 2 * hh];
      v8f acc = {0.f,0.f,0.f,0.f,0.f,0.f,0.f,0.f};
#pragma unroll
      for (int kb = 0; kb < 16; ++kb) acc = wmma_f32_k4(af[kb], bf[kb], acc);
      if (o < SIZE_M) {
#pragma unroll
        for (int j = 0; j < 8; ++j)
          sW[(mt * 16 + j + 8 * hh) * WST + o] = acc[j];
      }
    }
  }

  // ---- Phase 2b: e = sigmoid(v@eW^T + eb), a = tanh(v@aW^T + ab) ------
  for (int nt = 0; nt < 4; ++nt) {
    int o = nt * 16 + r16;
    v2f be[16], ba[16];
#pragma unroll
    for (int kb = 0; kb < 16; ++kb) {
      be[kb] = *(const v2f*)&eW[o * 64 + kb * 4 + 2 * hh];
      ba[kb] = *(const v2f*)&aW[o * 64 + kb * 4 + 2 * hh];
    }
    float ebv = eb[o], abv = ab[o];
    for (int mt = wave; mt < RP / 16; mt += 8) {
      v2f af[16];
#pragma unroll
      for (int kb = 0; kb < 16; ++kb)
        af[kb] = *(const v2f*)&sV[(mt * 16 + r16) * 64 + kb * 4 + 2 * hh];
      v8f ce = {0.f,0.f,0.f,0.f,0.f,0.f,0.f,0.f};
      v8f ca = {0.f,0.f,0.f,0.f,0.f,0.f,0.f,0.f};
#pragma unroll
      for (int kb = 0; kb < 16; ++kb) {
        ce = wmma_f32_k4(af[kb], be[kb], ce);
        ca = wmma_f32_k4(af[kb], ba[kb], ca);
      }
#pragma unroll
      for (int j = 0; j < 8; ++j) {
        int row = mt * 16 + j + 8 * hh;
        sE[row * 64 + o] = 1.f / (1.f + __expf(-(ce[j] + ebv)));
        sA[row * 64 + o] = tanhf(ca[j] + abv);
      }
    }
  }
  __syncthreads();

  // ---- Phase 3: row-wise softmax over 50 memory slots -----------------
  for (int r = tid; r < TT; r += 256) {
    float mx = -1e30f;
    for (int m = 0; m < SIZE_M; ++m) mx = fmaxf(mx, sW[r * WST + m]);
    float s = 0.f;
    for (int m = 0; m < SIZE_M; ++m) {
      float ev = __expf(sW[r * WST + m] - mx);
      sW[r * WST + m] = ev; s += ev;
    }
    float inv = 1.f / s;
    for (int m = 0; m < SIZE_M; ++m) sW[r * WST + m] *= inv;
  }
  __syncthreads();

  // ---- Phase 4: sequential scan over T (the latency-critical part) ----
  const int dd = tid & 63, gg = tid >> 6;
  for (int t = 0; t < TT; ++t) {
    // Mv update: Mv = Mv*(1 - w*e) + w*a
    for (int idx = tid; idx < SIZE_M * 64; idx += 256) {
      int m = idx >> 6, d = idx & 63;
      float wt = sW[t * WST + m];
      float et = sE[t * 64 + d];
      float at = sA[t * 64 + d];
      sMv[idx] = sMv[idx] * (1.f - wt * et) + wt * at;
    }
    __syncthreads();
    if (t < TT - 1) {
      // read[t][d] = sum_m w[t+1][m] * Mv[m][d]  (4-way split over m)
      float p = 0.f;
      for (int m = gg; m < SIZE_M; m += 4)
        p += sW[(t + 1) * WST + m] * sMv[m * 64 + dd];
      sRed[gg * 64 + dd] = p;
      __syncthreads();
      if (tid < 64)
        sV[t * 64 + tid] = sRed[tid] + sRed[64 + tid] + sRed[128 + tid] + sRed[192 + tid];
      __syncthreads();
    } else {
      __syncthreads();
    }
  }

  // zero read pad rows (199..207) before the f-GEMM
  for (int idx = (TT - 1) * 64 + tid; idx < RP * 64; idx += 256) sV[idx] = 0.f;
  __syncthreads();

  // ---- Phase 5: f = tanh([read, k(+1)] @ fW^T + fb), K=128 WMMA -------
  for (int nt = 0; nt < 4; ++nt) {
    int o = nt * 16 + r16;
    v2f bf[32];
#pragma unroll
    for (int kb = 0; kb < 32; ++kb)
      bf[kb] = *(const v2f*)&fW[o * 128 + kb * 4 + 2 * hh];
    float fbv = fb[o];
    for (int mt = wave; mt < RP / 16; mt += 8) {
      int row = mt * 16 + r16;
      v2f af[32];
#pragma unroll
      for (int kb = 0; kb < 16; ++kb)
        af[kb] = *(const v2f*)&sV[row * 64 + kb * 4 + 2 * hh];           // read half
#pragma unroll
      for (int kb = 16; kb < 32; ++kb)
        af[kb] = *(const v2f*)&sK[(row + 1) * 64 + (kb - 16) * 4 + 2 * hh]; // k[t'+1] half
      v8f acc = {0.f,0.f,0.f,0.f,0.f,0.f,0.f,0.f};
#pragma unroll
      for (int kb = 0; kb < 32; ++kb) acc = wmma_f32_k4(af[kb], bf[kb], acc);
#pragma unroll
      for (int j = 0; j < 8; ++j) {
        int rr = mt * 16 + j + 8 * hh;
        if (rr < TT - 1) sE[rr * 64 + o] = tanhf(acc[j] + fbv);          // sE reused as f
      }
    }
  }
  __syncthreads();

  // ---- Phase 6: heads + logits ----------------------------------------
  for (int r = tid; r < TT - 1; r += 256) {
    float s1 = 0.f, s2 = 0.f;
    for (int c = 0; c < 64; ++c) {
      s1 += sE[r * 64 + c] * abilW[c];
      s2 += sK[(r + 1) * 64 + c] * diffW[c];
    }
    float ability = tanhf(s1 + abilb[0]);
    float diff    = tanhf(s2 + diffb[0]);
    out[b * (TT - 1) + r] = 3.f * ability - diff;
  }
}

extern "C" void kernel_launch(void* const* d_in, const int* in_sizes, int n_in,
                              void* d_out, int out_size, void* d_ws, size_t ws_size,
                              hipStream_t stream) {
  (void)in_sizes; (void)n_in; (void)out_size; (void)d_ws; (void)ws_size;
  const int*   question = (const int*)  d_in[0];
  const int*   response = (const int*)  d_in[1];
  const float* vq_emb   = (const float*)d_in[2];
  const float* vc_emb   = (const float*)d_in[3];
  const float* kq_emb   = (const float*)d_in[4];
  const float* kc_emb   = (const float*)d_in[5];
  const float* Mk       = (const float*)d_in[6];
  const float* Mv0      = (const float*)d_in[7];
  const float* eW       = (const float*)d_in[8];
  const float* ebv      = (const float*)d_in[9];
  const float* aW       = (const float*)d_in[10];
  const float* abv      = (const float*)d_in[11];
  const float* fW       = (const float*)d_in[12];
  const float* fbv      = (const float*)d_in[13];
  const float* abilW    = (const float*)d_in[14];
  const float* abilb    = (const float*)d_in[15];
  const float* diffW    = (const float*)d_in[16];
  const float* diffb    = (const float*)d_in[17];
  float* out = (float*)d_out;

  const size_t smem_bytes =
      (size_t)(RK * 64 + 3 * RP * 64 + RP * WST + SIZE_M * 64 + 4 * 64) * sizeof(float)
      + (size_t)(2 * TT) * sizeof(int);   // ~266 KB of the WGP's 320 KB LDS

  (void)hipFuncSetAttribute((const void*)deepirt_fused_kernel,
                            hipFuncAttributeMaxDynamicSharedMemorySize,
                            (int)smem_bytes);

  deepirt_fused_kernel<<<BATCH, 256, smem_bytes, stream>>>(
      question, response, vq_emb, vc_emb, kq_emb, kc_emb, Mk, Mv0,
      eW, ebv, aW, abv, fW, fbv, abilW, abilb, diffW, diffb, out);
}